// RelationCrossing_86706799772304
// MI455X (gfx1250) — compile-verified
//
#include <hip/hip_runtime.h>
#include <stdint.h>

// Problem constants (from reference)
constexpr int NREL = 8;          // relations (softmax axis)
constexpr int NHEAD = 8;         // heads
constexpr int DIM = 64;          // per-head dim
constexpr int KD = NHEAD * DIM;  // 512 floats per (relation, node)
constexpr float NEG_SLOPE = 0.2f;

// Tiling
constexpr int THREADS = 256;                       // 8 wave32
constexpr int TILE_NODES = 2;                      // nodes per LDS stage
constexpr int TILES_PER_BLOCK = 8;                 // stages processed per block
constexpr int NODES_PER_BLOCK = TILE_NODES * TILES_PER_BLOCK;  // 16
constexpr int STAGE_FLOATS = TILE_NODES * NREL * KD;           // 8192 floats = 32 KB
constexpr int STAGE_BYTES = STAGE_FLOATS * 4;                  // 32 KB
constexpr int CHUNK_BYTES = TILE_NODES * KD * 4;               // 4 KB per relation chunk

typedef __attribute__((ext_vector_type(4))) float v4f;

// CDNA5 async DMA (SADDR form): global -> LDS, 16B per active lane, ASYNCcnt-tracked.
// mem = SGPR_base + VGPR_offset ; LDS = VGPR lds addr. Non-temporal: feats are
// streamed once (1.6 GB >> 192 MB L2), keep them out of L2.
__device__ __forceinline__ void async_copy_b128(uint32_t lds_byte_addr,
                                                uint64_t sbase, uint32_t voff) {
    asm volatile("global_load_async_to_lds_b128 %0, %1, %2 th:TH_LOAD_NT"
                 :
                 : "v"(lds_byte_addr), "v"(voff), "s"(sbase)
                 : "memory");
}

__global__ __launch_bounds__(THREADS)
void relation_crossing_kernel(const float* __restrict__ feats,
                              const float* __restrict__ rel_attn,
                              float* __restrict__ out,
                              int n_nodes) {
    __shared__ float smem[2 * STAGE_FLOATS];   // 64 KB, double buffered

    const int tid = threadIdx.x;
    const int block_n0 = blockIdx.x * NODES_PER_BLOCK;

    // Compute-phase mapping: 16 lanes per (node-in-tile, head)
    const int pair = tid >> 4;        // 0..15
    const int l16  = tid & 15;        // lane within group
    const int nn   = pair >> 3;       // node within tile: 0..1
    const int k    = pair & 7;        // head
    const v4f a4 = *reinterpret_cast<const v4f*>(rel_attn + k * DIM + l16 * 4);

    const uint32_t lds_base = (uint32_t)(uintptr_t)(&smem[0]);
    const uint32_t voff = (uint32_t)tid * 16u;           // per-lane byte offset
    const uint64_t gstep = (uint64_t)n_nodes * KD * 4u;  // relation stride (bytes)

    // One stage: 8 async b128 copies per thread. Packet index == tid, chunk == j,
    // so the chunk base is wave-uniform (SGPR) and the lane part is constant.
    auto issue_chunks = [&](int n0, int buf) {
        uint64_t g = (uint64_t)(uintptr_t)feats + (uint64_t)n0 * (KD * 4u);
        uint32_t lds_off = lds_base + (uint32_t)buf * STAGE_BYTES + voff;
#pragma unroll
        for (int j = 0; j < NREL; ++j) {
            async_copy_b128(lds_off, g, voff);
            g += gstep;
            lds_off += CHUNK_BYTES;
        }
    };
    auto issue_stage = [&](int tile, int buf) {
        const int n0 = block_n0 + tile * TILE_NODES;
        if (n0 + TILE_NODES <= n_nodes) {
            issue_chunks(n0, buf);                       // uniform fast path
        } else if (n0 + (tid >= (KD / 4) ? 1 : 0) < n_nodes) {
            issue_chunks(n0, buf);                       // divergent tail guard
        }
    };

    issue_stage(0, 0);

    for (int t = 0; t < TILES_PER_BLOCK; ++t) {
        const int buf = t & 1;
        if (t + 1 < TILES_PER_BLOCK) {
            const int next_n0 = block_n0 + (t + 1) * TILE_NODES;
            issue_stage(t + 1, (t + 1) & 1);
            if (next_n0 + TILE_NODES <= n_nodes) {
                // next stage issued exactly 8 per wave -> <=8 means stage t done
                asm volatile("s_wait_asynccnt 0x8" ::: "memory");
            } else {
                asm volatile("s_wait_asynccnt 0x0" ::: "memory");
            }
        } else {
            asm volatile("s_wait_asynccnt 0x0" ::: "memory");
        }
        __syncthreads();

        const int n = block_n0 + t * TILE_NODES + nn;
        const float* st = smem + buf * STAGE_FLOATS;

        // All 8 relations' float4 slices; partial dot against rel_attn
        v4f v[NREL];
        float s[NREL];
#pragma unroll
        for (int r = 0; r < NREL; ++r) {
            v[r] = *reinterpret_cast<const v4f*>(st + r * (TILE_NODES * KD) + nn * KD + k * DIM + l16 * 4);
            s[r] = v[r].x * a4.x + v[r].y * a4.y + v[r].z * a4.z + v[r].w * a4.w;
        }
        // Butterfly reduction across the 16 lanes of this (node, head) group
#pragma unroll
        for (int m = 1; m < 16; m <<= 1) {
#pragma unroll
            for (int r = 0; r < NREL; ++r) {
                s[r] += __shfl_xor(s[r], m, 16);
            }
        }
        // leaky_relu then softmax over relations (replicated across the 16 lanes)
        float mx = -INFINITY;
#pragma unroll
        for (int r = 0; r < NREL; ++r) {
            s[r] = fmaxf(s[r], 0.0f) + NEG_SLOPE * fminf(s[r], 0.0f);
            mx = fmaxf(mx, s[r]);
        }
        float sum = 0.0f;
#pragma unroll
        for (int r = 0; r < NREL; ++r) {
            s[r] = __expf(s[r] - mx);
            sum += s[r];
        }
        const float inv = 1.0f / sum;
        v4f o;
        o.x = 0.0f; o.y = 0.0f; o.z = 0.0f; o.w = 0.0f;
#pragma unroll
        for (int r = 0; r < NREL; ++r) {
            const float w = s[r] * inv;
            o.x += w * v[r].x;
            o.y += w * v[r].y;
            o.z += w * v[r].z;
            o.w += w * v[r].w;
        }
        if (n < n_nodes) {
            v4f* op = reinterpret_cast<v4f*>(out + (size_t)n * KD + k * DIM + l16 * 4);
            __builtin_nontemporal_store(o, op);   // streamed output, bypass L2 retention
        }
        __syncthreads();
    }
}

extern "C" void kernel_launch(void* const* d_in, const int* in_sizes, int n_in,
                              void* d_out, int out_size, void* d_ws, size_t ws_size,
                              hipStream_t stream) {
    (void)n_in; (void)out_size; (void)d_ws; (void)ws_size;
    const float* feats    = (const float*)d_in[0];
    const float* rel_attn = (const float*)d_in[1];
    float* out = (float*)d_out;

    const int n_nodes = in_sizes[0] / (NREL * KD);   // 100000
    const int blocks = (n_nodes + NODES_PER_BLOCK - 1) / NODES_PER_BLOCK;  // 6250

    relation_crossing_kernel<<<dim3(blocks), dim3(THREADS), 0, stream>>>(
        feats, rel_attn, out, n_nodes);
}